// GlueVAE_88467736363400
// MI455X (gfx1250) — compile-verified
//
#include <hip/hip_runtime.h>
#include <hip/hip_bf16.h>
#include <math.h>

// ---------------------------------------------------------------------------
// GlueVAE forward on gfx1250 (MI455X).
// Dense GEMMs: V_WMMA_F32_16X16X4_F32, 2x2 tiles per wave (32x32 output) for
// A/B fragment reuse. Edge message passing: coalesced global f32 atomics.
// Row reductions: LDS tree reductions (one 128-thread block per node).
// ---------------------------------------------------------------------------

#define NN   4096
#define EE   32768
#define GG   32
#define HH   128
#define H3   384
#define EDp  20      // ED=19 padded to multiple of 4
#define NG   16
#define RCUT 10.0f

typedef float v2f __attribute__((ext_vector_type(2)));
typedef float v8f __attribute__((ext_vector_type(8)));

// ---------------------------------------------------------------------------
// WMMA fp32 GEMM: C[M x Nout] = act(A[M x K] @ W[K x Nout] + b)
// One wave per 32x32 output tile (4 accumulators). K compile-time, K%4==0.
// A frag: lanes 0-15 -> rows, K pair (0,1); lanes 16-31 -> K pair (2,3).
// B frag: lane n holds col n; VGPR j holds B[kpair + j][n].
// C/D: VGPR j, lane-half hs -> m = j + hs*8, n = lane&15.
// ---------------------------------------------------------------------------
template <int K, int ACT>
__global__ __launch_bounds__(32) void k_gemm(const float* __restrict__ A,
                                             const float* __restrict__ W,
                                             const float* __restrict__ b,
                                             float* __restrict__ C, int Nout) {
  const int n0 = blockIdx.x * 32;
  const int m0 = blockIdx.y * 32;
  const int lane = threadIdx.x;
  const int lr = lane & 15;
  const int hs = lane >> 4;
  v8f acc00 = {}, acc01 = {}, acc10 = {}, acc11 = {};
  const float* a0 = A + (size_t)(m0 + lr) * K + hs * 2;        // rows m0..m0+15
  const float* a1 = a0 + (size_t)16 * K;                       // rows m0+16..m0+31
  const float* w0 = W + (size_t)(hs * 2) * Nout + n0 + lr;     // cols n0..n0+15
  const float* w1 = w0 + 16;                                   // cols n0+16..n0+31
  for (int k0 = 0; k0 < K; k0 += 4) {
    v2f af0 = *(const v2f*)(a0 + k0);   // 8B contiguous -> global_load_b64
    v2f af1 = *(const v2f*)(a1 + k0);
    const float* wp0 = w0 + (size_t)k0 * Nout;
    const float* wp1 = w1 + (size_t)k0 * Nout;
    v2f bf0, bf1;
    bf0.x = wp0[0];
    bf0.y = wp0[Nout];
    bf1.x = wp1[0];
    bf1.y = wp1[Nout];
    acc00 = __builtin_amdgcn_wmma_f32_16x16x4_f32(false, af0, false, bf0,
                                                  (short)0, acc00, false, false);
    acc01 = __builtin_amdgcn_wmma_f32_16x16x4_f32(false, af0, false, bf1,
                                                  (short)0, acc01, false, false);
    acc10 = __builtin_amdgcn_wmma_f32_16x16x4_f32(false, af1, false, bf0,
                                                  (short)0, acc10, false, false);
    acc11 = __builtin_amdgcn_wmma_f32_16x16x4_f32(false, af1, false, bf1,
                                                  (short)0, acc11, false, false);
  }
  const int na = n0 + lr;
  const int nb = na + 16;
  const float biasa = b ? b[na] : 0.0f;
  const float biasb = b ? b[nb] : 0.0f;
#pragma unroll
  for (int j = 0; j < 8; ++j) {
    const int ma = m0 + hs * 8 + j;
    const int mb = ma + 16;
    float v00 = acc00[j] + biasa;
    float v01 = acc01[j] + biasb;
    float v10 = acc10[j] + biasa;
    float v11 = acc11[j] + biasb;
    if (ACT) {
      v00 = v00 / (1.0f + __expf(-v00)) * 1.0f, v00 = v00;  // silu below
      // recompute cleanly (compiler folds): silu(x) = x * sigmoid(x)
    }
    if (ACT) {
      float s00 = 1.0f / (1.0f + __expf(-(acc00[j] + biasa)));
      float s01 = 1.0f / (1.0f + __expf(-(acc01[j] + biasb)));
      float s10 = 1.0f / (1.0f + __expf(-(acc10[j] + biasa)));
      float s11 = 1.0f / (1.0f + __expf(-(acc11[j] + biasb)));
      v00 = (acc00[j] + biasa) * s00;
      v01 = (acc01[j] + biasb) * s01;
      v10 = (acc10[j] + biasa) * s10;
      v11 = (acc11[j] + biasb) * s11;
    }
    C[(size_t)ma * Nout + na] = v00;
    C[(size_t)ma * Nout + nb] = v01;
    C[(size_t)mb * Nout + na] = v10;
    C[(size_t)mb * Nout + nb] = v11;
  }
}

static void gemm(hipStream_t st, const float* A, const float* W, const float* b,
                 float* C, int M, int K, int Nout, int act) {
  dim3 grid(Nout / 32, M / 32);
#define GEMM_CASE(KK)                                              \
  case KK:                                                         \
    if (act) k_gemm<KK, 1><<<grid, 32, 0, st>>>(A, W, b, C, Nout); \
    else     k_gemm<KK, 0><<<grid, 32, 0, st>>>(A, W, b, C, Nout); \
    break;
  switch (K) {
    GEMM_CASE(20)
    GEMM_CASE(128)
    GEMM_CASE(256)
    GEMM_CASE(384)
    default: break;
  }
#undef GEMM_CASE
}

// ---------------------------------------------------------------------------
// Small helper kernels
// ---------------------------------------------------------------------------
__global__ void k_init_mins(int* mi1, int* mi2) {
  int i = threadIdx.x;
  if (i < GG) { mi1[i] = NN; mi2[i] = NN; }
}

__global__ void k_seg_min(const int* __restrict__ isl, const int* __restrict__ mif,
                          const int* __restrict__ batch, int* mi1, int* mi2) {
  int n = blockIdx.x * blockDim.x + threadIdx.x;
  if (n >= NN) return;
  int b = batch[n];
  if (isl[n] == 0 && mif[n] == 1) atomicMin(&mi1[b], n);
  if (isl[n] == 1 && mif[n] == 1) atomicMin(&mi2[b], n);
}

__global__ void k_masks(const float* __restrict__ pos, const int* __restrict__ batch,
                        const int* __restrict__ isl, const int* __restrict__ mi1,
                        const int* __restrict__ mi2, const float* __restrict__ nv1,
                        const float* __restrict__ nv2, int* mask1, int* mask2,
                        float* pos1, float* pos2) {
  int n = blockIdx.x * blockDim.x + threadIdx.x;
  if (n >= NN) return;
  int b = batch[n];
  float px = pos[n * 3 + 0], py = pos[n * 3 + 1], pz = pos[n * 3 + 2];
  int m1 = mi1[b], m2 = mi2[b];
  int c1 = min(max(m1, 0), NN - 1);
  int c2 = min(max(m2, 0), NN - 1);
  float d1x = px - pos[c1 * 3 + 0], d1y = py - pos[c1 * 3 + 1], d1z = pz - pos[c1 * 3 + 2];
  float d2x = px - pos[c2 * 3 + 0], d2y = py - pos[c2 * 3 + 1], d2z = pz - pos[c2 * 3 + 2];
  float d1 = sqrtf(d1x * d1x + d1y * d1y + d1z * d1z);
  float d2 = sqrtf(d2x * d2x + d2y * d2y + d2z * d2z);
  int recep = (isl[n] == 0), lig = (isl[n] == 1);
  int mk1 = (m1 < NN) && (d1 < RCUT) && recep;
  int mk2 = (m2 < NN) && (d2 < RCUT) && lig;
  mask1[n] = mk1;
  mask2[n] = mk2;
#pragma unroll
  for (int c = 0; c < 3; ++c) {
    pos1[n * 3 + c] = mk1 ? nv1[n * 3 + c] * 0.5f : pos[n * 3 + c];
    pos2[n * 3 + c] = mk2 ? nv2[n * 3 + c] * 0.5f : pos[n * 3 + c];
  }
}

__global__ void k_edge_attr(const float* __restrict__ posv, const int* __restrict__ ei,
                            const float* __restrict__ eattr, float* __restrict__ ea,
                            float* __restrict__ dirv) {
  int e = blockIdx.x * blockDim.x + threadIdx.x;
  if (e >= EE) return;
  int r = ei[e], c = ei[EE + e];
  float dx = posv[r * 3 + 0] - posv[c * 3 + 0];
  float dy = posv[r * 3 + 1] - posv[c * 3 + 1];
  float dz = posv[r * 3 + 2] - posv[c * 3 + 2];
  float dist = sqrtf(dx * dx + dy * dy + dz * dz + 1e-8f);
  dirv[e * 3 + 0] = dx / dist;
  dirv[e * 3 + 1] = dy / dist;
  dirv[e * 3 + 2] = dz / dist;
  float* row = ea + (size_t)e * EDp;
  row[0] = eattr[(size_t)e * 19 + 0];
  row[1] = eattr[(size_t)e * 19 + 1];
  row[2] = eattr[(size_t)e * 19 + 2];
  const float step = 10.0f / 15.0f;
  const float coeff = -1.125f;  // -0.5 / (10/15)^2
#pragma unroll
  for (int g = 0; g < NG; ++g) {
    float d = dist - step * (float)g;
    row[3 + g] = __expf(coeff * d * d);
  }
  row[19] = 0.0f;  // pad
}

__global__ void k_pad_eweight(const float* __restrict__ ew, float* __restrict__ out) {
  int idx = blockIdx.x * blockDim.x + threadIdx.x;
  if (idx >= EDp * H3) return;
  int r = idx / H3, c = idx % H3;
  out[idx] = (r < 19) ? ew[r * H3 + c] : 0.0f;
}

__global__ void k_emb(const int* __restrict__ z, const float* __restrict__ emb,
                      float* __restrict__ s) {
  int idx = blockIdx.x * blockDim.x + threadIdx.x;
  if (idx >= NN * HH) return;
  int n = idx / HH, h = idx % HH;
  s[idx] = emb[(size_t)z[n] * HH + h];
}

__global__ void k_emb_add(const int* __restrict__ z, const float* __restrict__ emb,
                          const float* __restrict__ s1, float* __restrict__ s) {
  int idx = blockIdx.x * blockDim.x + threadIdx.x;
  if (idx >= NN * HH) return;
  int n = idx / HH, h = idx % HH;
  s[idx] = emb[(size_t)z[n] * HH + h] + s1[idx];
}

__global__ void k_edge_message(const float* __restrict__ phi, const float* __restrict__ wf,
                               const int* __restrict__ ei, const float* __restrict__ v,
                               const float* __restrict__ dirv, float* __restrict__ dsum,
                               float* __restrict__ dvsum) {
  unsigned idx = blockIdx.x * blockDim.x + threadIdx.x;
  if (idx >= (unsigned)EE * HH) return;
  int e = idx / HH, h = idx % HH;
  int r = ei[e], c = ei[EE + e];
  const float* ph = phi + (size_t)c * H3;
  const float* w = wf + (size_t)e * H3;
  float ds  = ph[h]          * w[h];
  float dvv = ph[HH + h]     * w[HH + h];
  float dvs = ph[2 * HH + h] * w[2 * HH + h];
  atomicAdd(&dsum[(size_t)r * HH + h], ds);
  const float* vc = v + (size_t)c * 3 * HH + h;
  float dx = dirv[e * 3 + 0], dy = dirv[e * 3 + 1], dz = dirv[e * 3 + 2];
  atomicAdd(&dvsum[((size_t)r * 3 + 0) * HH + h], dvv * vc[0]       + dvs * dx);
  atomicAdd(&dvsum[((size_t)r * 3 + 1) * HH + h], dvv * vc[HH]      + dvs * dy);
  atomicAdd(&dvsum[((size_t)r * 3 + 2) * HH + h], dvv * vc[2 * HH]  + dvs * dz);
}

__global__ void k_apply_deltas(float* __restrict__ s, float* __restrict__ v,
                               const float* __restrict__ dsum,
                               const float* __restrict__ dvsum) {
  int idx = blockIdx.x * blockDim.x + threadIdx.x;
  if (idx >= NN * HH) return;
  int n = idx / HH, h = idx % HH;
  s[idx] += dsum[idx];
#pragma unroll
  for (int c = 0; c < 3; ++c) {
    size_t o = ((size_t)n * 3 + c) * HH + h;
    v[o] += dvsum[o];
  }
}

__global__ void k_cat_vn(const float* __restrict__ s, const float* __restrict__ vv,
                         float* __restrict__ cat) {
  int idx = blockIdx.x * blockDim.x + threadIdx.x;
  if (idx >= NN * HH) return;
  int n = idx / HH, h = idx % HH;
  float a0 = vv[((size_t)n * 3 + 0) * HH + h];
  float a1 = vv[((size_t)n * 3 + 1) * HH + h];
  float a2 = vv[((size_t)n * 3 + 2) * HH + h];
  float vn = sqrtf(a0 * a0 + a1 * a1 + a2 * a2 + 1e-8f);
  cat[(size_t)n * 256 + h] = s[idx];
  cat[(size_t)n * 256 + HH + h] = vn;
}

__global__ void k_update(float* __restrict__ s, float* __restrict__ v,
                         const float* __restrict__ a, const float* __restrict__ uv,
                         const float* __restrict__ vv) {
  int idx = blockIdx.x * blockDim.x + threadIdx.x;
  if (idx >= NN * HH) return;
  int n = idx / HH, h = idx % HH;
  size_t o0 = ((size_t)n * 3 + 0) * HH + h;
  size_t o1 = o0 + HH, o2 = o0 + 2 * HH;
  float u0 = uv[o0], u1 = uv[o1], u2 = uv[o2];
  float w0 = vv[o0], w1 = vv[o1], w2 = vv[o2];
  float scal = u0 * w0 + u1 * w1 + u2 * w2;
  const float* ar = a + (size_t)n * H3;
  float avv = ar[h], asv = ar[HH + h], ass = ar[2 * HH + h];
  s[idx] += ass + asv * scal;
  v[o0] += avv * u0;
  v[o1] += avv * u1;
  v[o2] += avv * u2;
}

// one block (128 threads) per node; LDS tree reduction
__global__ __launch_bounds__(128) void k_normalize(const float* __restrict__ zz,
                                                   float* __restrict__ zp) {
  __shared__ float red[128];
  int n = blockIdx.x, h = threadIdx.x;
  float v = zz[(size_t)n * HH + h];
  red[h] = v * v;
  __syncthreads();
  for (int s = 64; s > 0; s >>= 1) {
    if (h < s) red[h] += red[h + s];
    __syncthreads();
  }
  float nrm = fmaxf(sqrtf(red[0]), 1e-12f);
  zp[(size_t)n * HH + h] = v / nrm;
}

// one block (128 threads) per node; coalesced atomics
__global__ __launch_bounds__(128) void k_masked_accum(
    const float* __restrict__ zp, const int* __restrict__ isl, int want,
    const int* __restrict__ batch, float* __restrict__ gz, float* __restrict__ den) {
  int n = blockIdx.x, h = threadIdx.x;
  if (isl[n] != want) return;  // uniform across block
  int b = batch[n];
  atomicAdd(&gz[(size_t)b * HH + h], zp[(size_t)n * HH + h]);
  if (h == 0) atomicAdd(&den[b], 1.0f);
}

__global__ void k_masked_final(float* __restrict__ out, const float* __restrict__ gz,
                               const float* __restrict__ den) {
  int idx = blockIdx.x * blockDim.x + threadIdx.x;
  if (idx >= GG * HH) return;
  int g = idx / HH;
  float d = den[g];
  out[idx] = (d > 0.0f) ? gz[idx] / fmaxf(d, 1.0f) : 0.0f;
}

// one block (128 threads) per node; 3 dot products via LDS tree reduction
__global__ __launch_bounds__(128) void k_delta(const float* __restrict__ vd,
                                               const float* __restrict__ vproj,
                                               const float* __restrict__ pos1,
                                               float* __restrict__ out) {
  __shared__ float red[3][128];
  int n = blockIdx.x, h = threadIdx.x;
  float w = vproj[h];
  red[0][h] = vd[((size_t)n * 3 + 0) * HH + h] * w;
  red[1][h] = vd[((size_t)n * 3 + 1) * HH + h] * w;
  red[2][h] = vd[((size_t)n * 3 + 2) * HH + h] * w;
  __syncthreads();
  for (int s = 64; s > 0; s >>= 1) {
    if (h < s) {
      red[0][h] += red[0][h + s];
      red[1][h] += red[1][h + s];
      red[2][h] += red[2][h + s];
    }
    __syncthreads();
  }
  if (h < 3) out[(size_t)n * 3 + h] = pos1[(size_t)n * 3 + h] + red[h][0];
}

__global__ void k_mask_out(const int* __restrict__ mask1, float* __restrict__ out) {
  int n = blockIdx.x * blockDim.x + threadIdx.x;
  if (n >= NN) return;
  out[n] = mask1[n] ? 1.0f : 0.0f;
}

// ---------------------------------------------------------------------------
// Host orchestration
// ---------------------------------------------------------------------------
struct Lp {
  const float *U, *V, *eb, *ew, *m1b, *m1w, *m2b, *m2w, *u1b, *u1w, *u2b, *u2w;
};

static inline int cdiv(int a, int b) { return (a + b - 1) / b; }

static void painn(hipStream_t st, float* s, float* v, const float* ea,
                  const float* dirv, const int* ei, const Lp* L, int nl,
                  float* const* eWp, float* tmp, float* phi, float* wf,
                  float* dsum, float* dvsum, float* uv, float* vv, float* cat,
                  float* u1o, float* a) {
  hipMemsetAsync(v, 0, (size_t)NN * 3 * HH * sizeof(float), st);
  for (int l = 0; l < nl; ++l) {
    gemm(st, s, L[l].m1w, L[l].m1b, tmp, NN, 128, 128, 1);      // silu(m1(s))
    gemm(st, tmp, L[l].m2w, L[l].m2b, phi, NN, 128, 384, 0);    // phi
    gemm(st, ea, eWp[l], L[l].eb, wf, EE, 20, 384, 0);          // wf
    hipMemsetAsync(dsum, 0, (size_t)NN * HH * sizeof(float), st);
    hipMemsetAsync(dvsum, 0, (size_t)NN * 3 * HH * sizeof(float), st);
    k_edge_message<<<cdiv(EE * HH, 256), 256, 0, st>>>(phi, wf, ei, v, dirv, dsum, dvsum);
    k_apply_deltas<<<cdiv(NN * HH, 256), 256, 0, st>>>(s, v, dsum, dvsum);
    gemm(st, v, L[l].U, nullptr, uv, 3 * NN, 128, 128, 0);
    gemm(st, v, L[l].V, nullptr, vv, 3 * NN, 128, 128, 0);
    k_cat_vn<<<cdiv(NN * HH, 256), 256, 0, st>>>(s, vv, cat);
    gemm(st, cat, L[l].u1w, L[l].u1b, u1o, NN, 256, 128, 1);
    gemm(st, u1o, L[l].u2w, L[l].u2b, a, NN, 128, 384, 0);
    k_update<<<cdiv(NN * HH, 256), 256, 0, st>>>(s, v, a, uv, vv);
  }
}

extern "C" void kernel_launch(void* const* d_in, const int* in_sizes, int n_in,
                              void* d_out, int out_size, void* d_ws, size_t ws_size,
                              hipStream_t stream) {
  // ---- input pointer map ----
  // Top-level: setup_inputs() insertion order. Nested pytrees: JAX tree_leaves
  // (dict keys sorted: ASCII => U,V,e,m1,m2,u1,u2 per layer; 'b' before 'w').
  const int*   z        = (const int*)d_in[0];
  const int*   ei       = (const int*)d_in[2];
  const float* eattr    = (const float*)d_in[3];
  const float* pos      = (const float*)d_in[4];
  const int*   isl      = (const int*)d_in[6];
  const int*   mif      = (const int*)d_in[7];
  const int*   batch    = (const int*)d_in[8];
  const float* nv1      = (const float*)d_in[9];
  const float* nv2      = (const float*)d_in[10];
  const float* enc_emb  = (const float*)d_in[11];
  auto layer_at = [&](int base, int l) {
    Lp p;
    int i = base + 12 * l;
    p.U   = (const float*)d_in[i + 0];
    p.V   = (const float*)d_in[i + 1];
    p.eb  = (const float*)d_in[i + 2];
    p.ew  = (const float*)d_in[i + 3];
    p.m1b = (const float*)d_in[i + 4];
    p.m1w = (const float*)d_in[i + 5];
    p.m2b = (const float*)d_in[i + 6];
    p.m2w = (const float*)d_in[i + 7];
    p.u1b = (const float*)d_in[i + 8];
    p.u1w = (const float*)d_in[i + 9];
    p.u2b = (const float*)d_in[i + 10];
    p.u2w = (const float*)d_in[i + 11];
    return p;
  };
  Lp encL[6], decL[4];
  for (int l = 0; l < 6; ++l) encL[l] = layer_at(12, l);
  const float* p_l1b = (const float*)d_in[84];
  const float* p_l1w = (const float*)d_in[85];
  const float* p_l2b = (const float*)d_in[86];
  const float* p_l2w = (const float*)d_in[87];
  const float* p_l3b = (const float*)d_in[88];
  const float* p_l3w = (const float*)d_in[89];
  const float* dec_emb = (const float*)d_in[90];
  for (int l = 0; l < 4; ++l) decL[l] = layer_at(91, l);
  const float* vproj = (const float*)d_in[139];

  // ---- workspace bump allocator ----
  char* wsp = (char*)d_ws;
  size_t off = 0;
  auto allocf = [&](size_t n) -> float* {
    float* r = (float*)(wsp + off);
    off += ((n * sizeof(float) + 255) / 256) * 256;
    return r;
  };
  auto alloci = [&](size_t n) -> int* {
    int* r = (int*)(wsp + off);
    off += ((n * sizeof(int) + 255) / 256) * 256;
    return r;
  };

  float* eWpad[10];
  for (int i = 0; i < 10; ++i) eWpad[i] = allocf((size_t)EDp * H3);
  float* ea1   = allocf((size_t)EE * EDp);
  float* ea2   = allocf((size_t)EE * EDp);
  float* dirv1 = allocf((size_t)EE * 3);
  float* dirv2 = allocf((size_t)EE * 3);
  float* pos1  = allocf((size_t)NN * 3);
  float* pos2  = allocf((size_t)NN * 3);
  int*   mi1   = alloci(GG);
  int*   mi2   = alloci(GG);
  int*   mask1 = alloci(NN);
  int*   mask2 = alloci(NN);
  float* s1    = allocf((size_t)NN * HH);
  float* s2    = allocf((size_t)NN * HH);
  float* sdec  = allocf((size_t)NN * HH);
  float* vbuf  = allocf((size_t)NN * 3 * HH);
  float* tmp   = allocf((size_t)NN * HH);
  float* phi   = allocf((size_t)NN * H3);
  float* wf    = allocf((size_t)EE * H3);
  float* dsum  = allocf((size_t)NN * HH);
  float* dvsum = allocf((size_t)NN * 3 * HH);
  float* uv    = allocf((size_t)NN * 3 * HH);
  float* vv    = allocf((size_t)NN * 3 * HH);
  float* cat   = allocf((size_t)NN * 256);
  float* u1o   = allocf((size_t)NN * HH);
  float* abuf  = allocf((size_t)NN * H3);
  float* h1    = allocf((size_t)NN * HH);
  float* h2    = allocf((size_t)NN * HH);
  float* zz    = allocf((size_t)NN * HH);
  float* zp    = allocf((size_t)NN * HH);
  float* gzs1  = allocf((size_t)GG * HH);
  float* gzs2  = allocf((size_t)GG * HH);
  float* den1  = allocf(GG);
  float* den2  = allocf(GG);
  (void)ws_size; (void)in_sizes; (void)n_in; (void)out_size;

  float* out = (float*)d_out;
  float* out_gz1  = out;              // 32*128
  float* out_gz2  = out + 4096;       // 32*128
  float* out_pos  = out + 8192;       // 4096*3
  float* out_mask = out + 20480;      // 4096

  // ---- masks and noised positions ----
  k_init_mins<<<1, GG, 0, stream>>>(mi1, mi2);
  k_seg_min<<<cdiv(NN, 256), 256, 0, stream>>>(isl, mif, batch, mi1, mi2);
  k_masks<<<cdiv(NN, 256), 256, 0, stream>>>(pos, batch, isl, mi1, mi2, nv1, nv2,
                                             mask1, mask2, pos1, pos2);

  // ---- edge attributes (RBF) + direction vectors ----
  k_edge_attr<<<cdiv(EE, 256), 256, 0, stream>>>(pos1, ei, eattr, ea1, dirv1);
  k_edge_attr<<<cdiv(EE, 256), 256, 0, stream>>>(pos2, ei, eattr, ea2, dirv2);

  // ---- pad edge-feature weights (19 -> 20 rows) ----
  for (int l = 0; l < 6; ++l)
    k_pad_eweight<<<cdiv(EDp * H3, 256), 256, 0, stream>>>(encL[l].ew, eWpad[l]);
  for (int l = 0; l < 4; ++l)
    k_pad_eweight<<<cdiv(EDp * H3, 256), 256, 0, stream>>>(decL[l].ew, eWpad[6 + l]);

  // ---- encoder, view 1 ----
  k_emb<<<cdiv(NN * HH, 256), 256, 0, stream>>>(z, enc_emb, s1);
  painn(stream, s1, vbuf, ea1, dirv1, ei, encL, 6, &eWpad[0],
        tmp, phi, wf, dsum, dvsum, uv, vv, cat, u1o, abuf);

  // ---- encoder, view 2 ----
  k_emb<<<cdiv(NN * HH, 256), 256, 0, stream>>>(z, enc_emb, s2);
  painn(stream, s2, vbuf, ea2, dirv2, ei, encL, 6, &eWpad[0],
        tmp, phi, wf, dsum, dvsum, uv, vv, cat, u1o, abuf);

  // ---- projection heads + masked means ----
  hipMemsetAsync(gzs1, 0, (size_t)GG * HH * sizeof(float), stream);
  hipMemsetAsync(gzs2, 0, (size_t)GG * HH * sizeof(float), stream);
  hipMemsetAsync(den1, 0, GG * sizeof(float), stream);
  hipMemsetAsync(den2, 0, GG * sizeof(float), stream);

  gemm(stream, s1, p_l1w, p_l1b, h1, NN, 128, 128, 1);
  gemm(stream, h1, p_l2w, p_l2b, h2, NN, 128, 128, 1);
  gemm(stream, h2, p_l3w, p_l3b, zz, NN, 128, 128, 0);
  k_normalize<<<NN, 128, 0, stream>>>(zz, zp);
  k_masked_accum<<<NN, 128, 0, stream>>>(zp, isl, 1, batch, gzs1, den1);

  gemm(stream, s2, p_l1w, p_l1b, h1, NN, 128, 128, 1);
  gemm(stream, h1, p_l2w, p_l2b, h2, NN, 128, 128, 1);
  gemm(stream, h2, p_l3w, p_l3b, zz, NN, 128, 128, 0);
  k_normalize<<<NN, 128, 0, stream>>>(zz, zp);
  k_masked_accum<<<NN, 128, 0, stream>>>(zp, isl, 0, batch, gzs2, den2);

  k_masked_final<<<cdiv(GG * HH, 256), 256, 0, stream>>>(out_gz1, gzs1, den1);
  k_masked_final<<<cdiv(GG * HH, 256), 256, 0, stream>>>(out_gz2, gzs2, den2);

  // ---- decoder (view 1 geometry) ----
  k_emb_add<<<cdiv(NN * HH, 256), 256, 0, stream>>>(z, dec_emb, s1, sdec);
  painn(stream, sdec, vbuf, ea1, dirv1, ei, decL, 4, &eWpad[6],
        tmp, phi, wf, dsum, dvsum, uv, vv, cat, u1o, abuf);

  // ---- outputs ----
  k_delta<<<NN, 128, 0, stream>>>(vbuf, vproj, pos1, out_pos);
  k_mask_out<<<cdiv(NN, 256), 256, 0, stream>>>(mask1, out_mask);
}